// HNet_78915729096799
// MI455X (gfx1250) — compile-verified
//
#include <hip/hip_runtime.h>
#include <hip/hip_bf16.h>
#include <stdint.h>

// ---------------------------------------------------------------------------
// HNet dechunk (EMA plug-back) for MI455X / gfx1250.
// B=4, L=8192, D=1024 fp32.  Bandwidth bound (~260MB => ~11us @ 23.3 TB/s).
// Chunked-scan: 16 serial EMA steps -> one 16x16 matmul via
// V_WMMA_F32_16X16X4_F32; global->LDS staging via async path
// (global_load_async_to_lds_b128 + s_wait_asynccnt).
// Rev 3: the M-matrix (EMA transition weights) is factored as
//   M[m][j] = cumA[m] * (p_j / cumA[j])   (j <= m)
// with cumA/inv precomputed per 16-block into small global arrays (K1b),
// eliminating the per-lane predicated running product that previously
// lowered to v_cmp/v_cndmask select chains in K3.
// ---------------------------------------------------------------------------

constexpr int   Bn     = 4;
constexpr int   Ln     = 8192;
constexpr int   Dn     = 1024;
constexpr int   CH     = 128;         // timesteps per chunk
constexpr int   NC     = Ln / CH;     // 64 chunks
constexpr int   DSPLIT = 4;           // D split for K3 (256 cols per WG)
constexpr int   DEPTH  = 8;           // async ring depth (rows)
constexpr float EPSC   = 1e-4f;

typedef __attribute__((ext_vector_type(2))) float v2f;
typedef __attribute__((ext_vector_type(8))) float v8f;
typedef __attribute__((address_space(3))) float lds_f;
typedef const __attribute__((address_space(1))) float glob_cf;

__device__ __forceinline__ void async_copy_b128(lds_f* lp, glob_cf* gp) {
  // CDNA5 async DMA: copies 16 bytes per active lane from the global address in gp to the LDS address in lp, tracked by ASYNCcnt.
  asm volatile("global_load_async_to_lds_b128 %0, %1, off" :: "v"(lp), "v"(gp) : "memory");
}

// ---------------------------------------------------------------------------
// K1: per-batch boundary scan.
//   pcomp[b][k] = clipped p at k-th boundary (0 for k>=K)
//   pos[b][k]   = position of k-th boundary; pos[b][K..L] = L
//   Kcnt[b]     = number of boundaries
// ---------------------------------------------------------------------------
__global__ __launch_bounds__(1024) void k1_scan(
    const unsigned char* __restrict__ mask,
    const float* __restrict__ bprob,
    float* __restrict__ pcomp, int* __restrict__ pos, int* __restrict__ Kcnt) {
  const int b = blockIdx.x, tid = threadIdx.x;
  __shared__ int sdata[1024];
  const int base = tid * 8;
  int mloc[8];
  int s = 0;
  #pragma unroll
  for (int j = 0; j < 8; ++j) {
    int l = base + j;
    int mm = mask[(size_t)b * Ln + l] ? 1 : 0;
    if (l == 0) mm = 1;            // first token always a boundary
    mloc[j] = mm; s += mm;
  }
  sdata[tid] = s;
  __syncthreads();
  for (int off = 1; off < 1024; off <<= 1) {      // Hillis-Steele inclusive scan
    int v = (tid >= off) ? sdata[tid - off] : 0;
    __syncthreads();
    if (tid >= off) sdata[tid] += v;
    __syncthreads();
  }
  int cum = sdata[tid] - s;                        // exclusive prefix
  #pragma unroll
  for (int j = 0; j < 8; ++j) {
    int l = base + j;
    cum += mloc[j];
    if (mloc[j]) {
      int k = cum - 1;
      pos[(size_t)b * (Ln + 1) + k] = l;
      float p = bprob[((size_t)b * Ln + l) * 2 + 1];
      p = fminf(fmaxf(p, EPSC), 1.0f - EPSC);
      pcomp[(size_t)b * Ln + k] = p;
    }
  }
  int K = sdata[1023];
  if (tid == 0) Kcnt[b] = K;
  for (int k = K + tid; k <= Ln; k += 1024) pos[(size_t)b * (Ln + 1) + k] = Ln;
  for (int k = K + tid; k < Ln; k += 1024) pcomp[(size_t)b * Ln + k] = 0.0f;
}

// ---------------------------------------------------------------------------
// K1b: per 16-step block factors (straight-line, lane-independent):
//   cum16[k] = prod_{i=blk..k} (1-p_i)          (blk = k & ~15)
//   inv16[k] = p_k / cum16[k]                   (so M[m][j] = cum16[m]*inv16[j])
// ---------------------------------------------------------------------------
__global__ __launch_bounds__(512) void k1b_blockfac(
    const float* __restrict__ pcomp,
    float* __restrict__ cum16, float* __restrict__ inv16) {
  const int b = blockIdx.x;
  const size_t base = (size_t)b * Ln + (size_t)threadIdx.x * 16;
  float cv[16], iv[16];
  float run = 1.0f;
  #pragma unroll
  for (int i = 0; i < 16; ++i) {
    float p = pcomp[base + i];
    run *= (1.0f - p);
    cv[i] = run;
    iv[i] = p / fmaxf(run, 1e-30f);
  }
  #pragma unroll
  for (int q = 0; q < 4; ++q) {
    float4 a; a.x = cv[4*q]; a.y = cv[4*q+1]; a.z = cv[4*q+2]; a.w = cv[4*q+3];
    float4 c; c.x = iv[4*q]; c.y = iv[4*q+1]; c.z = iv[4*q+2]; c.w = iv[4*q+3];
    ((float4*)(cum16 + base))[q] = a;
    ((float4*)(inv16 + base))[q] = c;
  }
}

// ---------------------------------------------------------------------------
// K2: per (batch, chunk): chunk product W_c and chunk-final state
//   u_c[d] = sum_j coef_j * x[j][d],  coef_j = p_j * prod_{i>j}(1-p_i).
// x rows staged via async-to-LDS ring (each lane consumes its own 16B).
// ---------------------------------------------------------------------------
__global__ __launch_bounds__(256) void k2_chunk(
    const float* __restrict__ x, const float* __restrict__ pcomp,
    const int* __restrict__ Kcnt,
    float* __restrict__ u, float* __restrict__ Wg) {
  const int wg = blockIdx.x;
  const int b = wg / NC, c = wg % NC;
  const int tid = threadIdx.x;                     // 256 threads, 4 channels each
  const int c0 = c * CH;
  const int Kb = Kcnt[b];
  const size_t ubase = ((size_t)(b * NC + c)) * Dn;

  if (c0 >= Kb) {                                  // chunk past last boundary
    float4 z = {0.f, 0.f, 0.f, 0.f};
    ((float4*)(u + ubase))[tid] = z;
    if (tid == 0) Wg[b * NC + c] = 1.0f;
    return;
  }

  __shared__ float scoef[CH];
  __shared__ float sx[DEPTH][Dn];

  if (tid == 0) {                                  // suffix products (128 serial)
    float suf = 1.0f;
    for (int j = CH - 1; j >= 0; --j) {
      float pj = pcomp[(size_t)b * Ln + c0 + j];
      scoef[j] = pj * suf;
      suf *= (1.0f - pj);
    }
    Wg[b * NC + c] = suf;
  }
  __syncthreads();

  const float* xrow0 = x + ((size_t)b * Ln + c0) * Dn + tid * 4;
  #pragma unroll
  for (int j = 0; j < DEPTH; ++j)
    async_copy_b128((lds_f*)&sx[j][tid * 4], (glob_cf*)(xrow0 + (size_t)j * Dn));

  float4 acc = {0.f, 0.f, 0.f, 0.f};
  for (int j = 0; j < CH; ++j) {
    if (j >= CH - DEPTH) { asm volatile("s_wait_asynccnt 0" ::: "memory"); }
    else                 { asm volatile("s_wait_asynccnt 7" ::: "memory"); }
    const float4 xv = *(const float4*)&sx[j % DEPTH][tid * 4];
    const float cf = scoef[j];
    acc.x += cf * xv.x; acc.y += cf * xv.y; acc.z += cf * xv.z; acc.w += cf * xv.w;
    if (j + DEPTH < CH) {
      asm volatile("s_wait_dscnt 0" ::: "memory");   // LDS reads done before ring reuse
      async_copy_b128((lds_f*)&sx[j % DEPTH][tid * 4],
                      (glob_cf*)(xrow0 + (size_t)(j + DEPTH) * Dn));
    }
  }
  ((float4*)(u + ubase))[tid] = acc;
}

// ---------------------------------------------------------------------------
// K2b: cross-chunk carry scan: carryIn[c] = W_{c-1}*carryIn[c-1] + u_{c-1}.
// ---------------------------------------------------------------------------
__global__ __launch_bounds__(1024) void k2b_carry(
    const float* __restrict__ Wg, const float* __restrict__ u,
    float* __restrict__ carryIn) {
  const int b = blockIdx.x, d = threadIdx.x;
  float acc = 0.0f;
  for (int c = 0; c < NC; ++c) {
    const size_t base = ((size_t)(b * NC + c)) * Dn;
    carryIn[base + d] = acc;
    acc = Wg[b * NC + c] * acc + u[base + d];
  }
}

// ---------------------------------------------------------------------------
// K3: chunk scan via WMMA.  For each 16-step block:
//   H(16x16) = M(16x16) @ X(16x16) + cumA(16) (x) carry(16)
// as 4 accumulating V_WMMA_F32_16X16X4_F32; carry chained through blocks;
// each h_k scattered over its output run [pos_k, pos_{k+1}).
// M entries come from cum16/inv16 loads (no lane-variant register indexing).
// Fragment layouts per CDNA5 ISA 7.12.2 (A 16x4, C/D 16x16 f32).
// ---------------------------------------------------------------------------
__global__ __launch_bounds__(128) void k3_wmma(
    const float* __restrict__ x,
    const float* __restrict__ cum16, const float* __restrict__ inv16,
    const int* __restrict__ pos, const int* __restrict__ Kcnt,
    const float* __restrict__ carryIn, float* __restrict__ out) {
  const int wg  = blockIdx.x;
  const int b   = wg / (NC * DSPLIT);
  const int rem = wg % (NC * DSPLIT);
  const int c   = rem / DSPLIT;
  const int ds  = rem % DSPLIT;
  const int wave = threadIdx.x >> 5;
  const int lane = threadIdx.x & 31;
  const int halfi = (lane >> 4) & 1;
  const int m    = lane & 15;

  const int Kb = Kcnt[b];
  if (c * CH >= Kb) return;                        // nothing plugged back here

  const int* posb = pos + (size_t)b * (Ln + 1);
  int   colb[4];
  float carryv[4];
  #pragma unroll
  for (int i = 0; i < 4; ++i) {
    colb[i] = ds * 256 + wave * 64 + i * 16;
    carryv[i] = carryIn[((size_t)(b * NC + c)) * Dn + colb[i] + m];
  }

  for (int t = 0; t < CH / 16; ++t) {
    const int kbase = c * CH + t * 16;
    if (kbase >= Kb) break;

    const float* crow = cum16 + (size_t)b * Ln + kbase;
    const float* irow = inv16 + (size_t)b * Ln + kbase;

    // cumA for this lane's A-row and for its 8 C/D rows (pure loads).
    const float cumAm = crow[m];
    float cumAv[8];
    #pragma unroll
    for (int v = 0; v < 8; ++v) cumAv[v] = crow[v + 8 * halfi];

    // A fragments: afr[s] = M[m][j], j = 4s+2*half (+1).
    // M[m][j] = cumA[m]*inv[j] for j<=m else 0  -> 1 load + 1 mul + 1 cndmask.
    v2f afr[4];
    #pragma unroll
    for (int s = 0; s < 4; ++s) {
      const int j0 = 4 * s + 2 * halfi;
      const float e0 = cumAm * irow[j0];
      const float e1 = cumAm * irow[j0 + 1];
      afr[s].x = (j0     <= m) ? e0 : 0.0f;
      afr[s].y = (j0 + 1 <= m) ? e1 : 0.0f;
    }

    const float* xrow = x + ((size_t)b * Ln + kbase) * Dn;
    #pragma unroll
    for (int i = 0; i < 4; ++i) {
      const int col = colb[i] + m;
      v8f cf = {};
      #pragma unroll
      for (int s = 0; s < 4; ++s) {
        const int r = 4 * s + 2 * halfi;          // address math only (vmem)
        v2f bfr;
        bfr.x = xrow[(size_t)r * Dn + col];
        bfr.y = xrow[(size_t)(r + 1) * Dn + col];
        cf = __builtin_amdgcn_wmma_f32_16x16x4_f32(
            false, afr[s], false, bfr, (short)0, cf, false, false);
      }
      // rank-1 carry term: H[r][n] += cumA_r * carry[n]; lane rows r = v+8*half.
      const float cO = carryv[i];
      #pragma unroll
      for (int v = 0; v < 8; ++v) cf[v] += cumAv[v] * cO;
      // next carry = H[15][n], held in c[7] of lane 16+n.
      carryv[i] = __int_as_float(
          __builtin_amdgcn_ds_bpermute((16 + m) << 2, __float_as_int(cf[7])));
      // scatter h_k over its run of output rows.
      #pragma unroll
      for (int v = 0; v < 8; ++v) {
        const int kg = kbase + v + 8 * halfi;     // address math only
        const int r0 = posb[kg], r1 = posb[kg + 1];
        const float hv = cf[v];
        float* op = out + (size_t)b * Ln * Dn + col;
        for (int l = r0; l < r1; ++l) op[(size_t)l * Dn] = hv;
      }
    }
  }
}

// ---------------------------------------------------------------------------
extern "C" void kernel_launch(void* const* d_in, const int* in_sizes, int n_in,
                              void* d_out, int out_size, void* d_ws, size_t ws_size,
                              hipStream_t stream) {
  (void)in_sizes; (void)n_in; (void)out_size; (void)ws_size;
  const float*         x     = (const float*)d_in[0];
  const float*         bprob = (const float*)d_in[1];
  const unsigned char* mask  = (const unsigned char*)d_in[2];  // jax bool -> u8
  float*               out   = (float*)d_out;

  char* ws = (char*)d_ws;
  size_t off = 0;
  float* pcomp = (float*)(ws + off); off += (size_t)Bn * Ln * 4;        // 128 KB
  float* cum16 = (float*)(ws + off); off += (size_t)Bn * Ln * 4;        // 128 KB
  float* inv16 = (float*)(ws + off); off += (size_t)Bn * Ln * 4;        // 128 KB
  int*   pos   = (int*)(ws + off);   off += (size_t)Bn * (Ln + 1) * 4;  // 128 KB
  int*   Kcnt  = (int*)(ws + off);   off += 16;                          // 16 B
  float* Wg    = (float*)(ws + off); off += (size_t)Bn * NC * 4;        // 1 KB
  float* u     = (float*)(ws + off); off += (size_t)Bn * NC * Dn * 4;   // 1 MB
  float* carry = (float*)(ws + off); off += (size_t)Bn * NC * Dn * 4;   // 1 MB

  k1_scan    <<<Bn,                1024, 0, stream>>>(mask, bprob, pcomp, pos, Kcnt);
  k1b_blockfac<<<Bn,                512, 0, stream>>>(pcomp, cum16, inv16);
  k2_chunk   <<<Bn * NC,            256, 0, stream>>>(x, pcomp, Kcnt, u, Wg);
  k2b_carry  <<<Bn,                1024, 0, stream>>>(Wg, u, carry);
  k3_wmma    <<<Bn * NC * DSPLIT,   128, 0, stream>>>(x, cum16, inv16, pos, Kcnt, carry, out);
}